// AdjunctionModelV3_38989713113333
// MI455X (gfx1250) — compile-verified
//
#include <hip/hip_runtime.h>
#include <hip/hip_bf16.h>
#include <math.h>

// ---------------------------------------------------------------------------
// Chamfer coherence on MI455X (gfx1250, wave32).
//   d2(i,j) = |a_i|^2 + (-2 a_i . b_j + |b_j|^2)
// One V_WMMA_F32_16X16X4_F32 per 16x16 tile:
//   A (16x4 f32): rows = orig points, K = {x, y, z, 1}
//   B (4x16 f32): cols = recon points, K = {-2x, -2y, -2z, |b|^2}
//   C (16x16)   : |a_i|^2 (loop-invariant)  ->  D = d2 directly.
// B operand pre-swizzled into LDS so the tile loop needs one ds_load_b64.
// All min reductions run on d2 *bit patterns* with signed-int min:
// for finite floats this matches float order on non-negatives and only
// mis-orders among negatives, which are clamped to 0 afterwards anyway, so
// max(int_min_result, 0) == max(true_min, 0) exactly. This avoids all
// v_max canonicalize ops and uses v_min_i32 / ds_min_i32 / atomic_min_i32.
// sqrt deferred through the min (monotone).
// ---------------------------------------------------------------------------

#define BATCHES 16
#define NPTS    4096
#define CHUNK   2048     // recon points staged per pass: 2*CHUNK float2 = 32 KB
#define INF_BITS 0x7F800000

typedef float v2f __attribute__((ext_vector_type(2)));
typedef float v8f __attribute__((ext_vector_type(8)));

__device__ __forceinline__ int imin2(int a, int b) { return a < b ? a : b; }

// ------------------------- init backward min buffer -------------------------
__global__ void chamfer_init(int* __restrict__ backBits) {
    int idx = blockIdx.x * 256 + threadIdx.x;          // 65536 total
    backBits[idx] = INF_BITS;
}

// ------------------------------ main kernel --------------------------------
// grid: (32 row-blocks, 16 batches), block: 256 threads (8 waves).
// Each wave owns a 16-row A tile; WG covers 128 rows x all 4096 columns.
__global__ void __launch_bounds__(256)
chamfer_main(const float* __restrict__ pos,
             const float* __restrict__ recon,
             float* __restrict__ fwdD2,            /* d_out + 16, gets d2  */
             int* __restrict__ backBits)           /* B*NPTS i32 in d_ws   */
{
    // sPQ[half][n] = float2 WMMA-B operand for lane-half `half`, column n:
    //   half 0 -> {-2x, -2y} (K rows 0,1)   half 1 -> {-2z, |b|^2} (K rows 2,3)
    // +16 pad keeps the software-pipeline preload in bounds.
    __shared__ float2 sPQ[2 * CHUNK + 16];           // 32 KB (+128 B)
    __shared__ int sColMin[NPTS];                    // 16 KB

    const int b    = blockIdx.y;
    const int tid  = threadIdx.x;
    const int lane = tid & 31;
    const int wave = tid >> 5;
    const int m    = lane & 15;                      // 0..15 (column / row id)
    const int half = lane >> 4;                      // 0 or 1

    // init per-WG column-min table
    for (int i = tid; i < NPTS; i += 256) sColMin[i] = INF_BITS;

    // ---- A tile (fixed for whole kernel): 16 rows of orig points ----
    const int rowBase = blockIdx.x * 128 + wave * 16;
    const size_t aIdx = ((size_t)b * NPTS + rowBase + m) * 3;
    const float ax = pos[aIdx + 0];
    const float ay = pos[aIdx + 1];
    const float az = pos[aIdx + 2];
    const float a2m = ax * ax + ay * ay + az * az;   // |a_(rowBase+m)|^2

    v2f A;
    A.x = half ? az   : ax;                          // K=0 / K=2
    A.y = half ? 1.0f : ay;                          // K=1 / K=3 (picks |b|^2)

    // Loop-invariant C: C[r] = |a|^2 of row M = rowBase + r + 8*half,
    // held by lane (r + 8*half).
    v8f Cfix;
#pragma unroll
    for (int r = 0; r < 8; ++r) Cfix[r] = __shfl(a2m, r + 8 * half, 32);

    int rowMin[8];
#pragma unroll
    for (int r = 0; r < 8; ++r) rowMin[r] = INF_BITS;

    for (int chunk = 0; chunk < NPTS; chunk += CHUNK) {
        __syncthreads();
        // ---- stage recon chunk into LDS, pre-swizzled as WMMA B operand ----
        for (int i = tid; i < CHUNK; i += 256) {
            const size_t gi = ((size_t)b * NPTS + chunk + i) * 3;
            const float bx = recon[gi + 0];
            const float by = recon[gi + 1];
            const float bz = recon[gi + 2];
            const float b2 = bx * bx + by * by + bz * bz;
            sPQ[i]         = make_float2(-2.0f * bx, -2.0f * by);
            sPQ[CHUNK + i] = make_float2(-2.0f * bz, b2);
            if (chunk + CHUNK < NPTS)                 // warm caches for pass 2
                __builtin_prefetch(&recon[gi + (size_t)CHUNK * 3], 0, 1);
        }
        __syncthreads();

        // ---- sweep 16-column tiles: 1 ds_load_b64 + 1 WMMA per tile, with
        //      a one-tile preload so the LDS load hides under the epilogue ---
        const float2* pq = &sPQ[half * CHUNK + m];
        float2 cur = pq[0];
#pragma unroll 4
        for (int t = 0; t < CHUNK; t += 16) {
            const float2 nxt = pq[t + 16];           // preload (padded array)

            v2f Bm;
            Bm.x = cur.x;
            Bm.y = cur.y;

            // D = A x B + |a|^2  ==  squared distance tile
            v8f D = __builtin_amdgcn_wmma_f32_16x16x4_f32(
                false, A, false, Bm, (short)0, Cfix, false, false);

            int cmin;
#pragma unroll
            for (int r = 0; r < 8; ++r) {
                const int d2b = __float_as_int(D[r]);
                rowMin[r] = imin2(rowMin[r], d2b);
                cmin = (r == 0) ? d2b : imin2(cmin, d2b);
            }
            // both halves hold an 8-row partial min for column chunk+t+m;
            // the LDS atomic unit merges them.
            atomicMin(&sColMin[chunk + t + m], cmin);

            cur = nxt;
        }
    }

    // ---- forward (row) mins: reduce across the 16 lanes sharing a row set --
    if (m == 0) {                                    // lanes 0 and 16 write
        // (shuffles executed by all lanes below; writes by lanes 0/16 only)
    }
#pragma unroll
    for (int r = 0; r < 8; ++r) {
        int v = rowMin[r];
        v = imin2(v, __shfl_xor(v, 1, 32));
        v = imin2(v, __shfl_xor(v, 2, 32));
        v = imin2(v, __shfl_xor(v, 4, 32));
        v = imin2(v, __shfl_xor(v, 8, 32));
        rowMin[r] = v;
    }
    if (m == 0) {
        const size_t base = (size_t)b * NPTS + rowBase + 8 * half;
#pragma unroll
        for (int r = 0; r < 8; ++r)                  // clamp once, store d2
            fwdD2[base + r] = fmaxf(__int_as_float(rowMin[r]), 0.0f);
    }

    // ---- flush backward (column) mins to global ----
    __syncthreads();
    for (int i = tid; i < NPTS; i += 256)
        atomicMin(&backBits[(size_t)b * NPTS + i], sColMin[i]);
}

// ----------------------------- finalize ------------------------------------
// grid: 16 (one block per batch), block: 256.
__global__ void __launch_bounds__(256)
chamfer_finalize(float* __restrict__ outScalar,        /* d_out[0..15]        */
                 float* __restrict__ spatial,          /* d_out+16, d2 -> dist */
                 const int* __restrict__ backBits)
{
    __shared__ float redF[256];
    __shared__ float redB[256];
    const int b = blockIdx.x;
    const int tid = threadIdx.x;

    float sf = 0.0f, sb = 0.0f;
    for (int j = tid; j < NPTS; j += 256) {
        const size_t idx = (size_t)b * NPTS + j;
        const float df = sqrtf(spatial[idx]);          // d2 (clamped) -> dist
        spatial[idx] = df;                             // coherence_spatial
        sf += df;
        sb += sqrtf(fmaxf(__int_as_float(backBits[idx]), 0.0f));
    }
    redF[tid] = sf;
    redB[tid] = sb;
    __syncthreads();
    for (int s = 128; s > 0; s >>= 1) {
        if (tid < s) {
            redF[tid] += redF[tid + s];
            redB[tid] += redB[tid + s];
        }
        __syncthreads();
    }
    if (tid == 0)
        outScalar[b] = (redF[0] + redB[0]) * (0.5f / (float)NPTS);
}

// ------------------------------ launcher -----------------------------------
extern "C" void kernel_launch(void* const* d_in, const int* in_sizes, int n_in,
                              void* d_out, int out_size, void* d_ws, size_t ws_size,
                              hipStream_t stream) {
    (void)in_sizes; (void)n_in; (void)out_size; (void)ws_size;
    const float* pos   = (const float*)d_in[0];
    const float* recon = (const float*)d_in[1];
    // d_in[2] (batch vector) unused: sorted equal-size segments, like reference.

    float* out       = (float*)d_out;
    float* outScalar = out;              // 16 floats
    float* spatial   = out + BATCHES;    // 65536 floats (holds d2, then dist)
    int* backBits    = (int*)d_ws;       // B*NPTS i32 = 256 KB

    chamfer_init<<<(BATCHES * NPTS) / 256, 256, 0, stream>>>(backBits);
    chamfer_main<<<dim3(NPTS / 128, BATCHES), 256, 0, stream>>>(
        pos, recon, spatial, backBits);
    chamfer_finalize<<<BATCHES, 256, 0, stream>>>(outScalar, spatial, backBits);
}